// Net_360777253060
// MI455X (gfx1250) — compile-verified
//
#include <hip/hip_runtime.h>
#include <math.h>

typedef float v2f __attribute__((ext_vector_type(2)));
typedef float v8f __attribute__((ext_vector_type(8)));

#define IN_F 64
#define HID  128
#define KCL  16

// agg[i*HID + c] = bias[c]  (scatter target pre-seeded with bias)
__global__ void k_init_bias(float* __restrict__ agg, const float* __restrict__ b, int total) {
    int i = blockIdx.x * blockDim.x + threadIdx.x;
    if (i < total) agg[i] = b[i & (HID - 1)];
}

// C[N x NO] = (RELU ? max(A,0) : A)[N x K] @ B[K x NO], fp32 WMMA 16x16x4.
// One wave computes one 16x16 tile. Requires N%16==0, K%4==0, NO%16==0 (true here),
// so EXEC is all-ones for every wave (WMMA requirement).
template <bool RELU>
__global__ void k_gemm_wmma(const float* __restrict__ A, const float* __restrict__ B,
                            float* __restrict__ C, int K, int NO) {
    const int lane = threadIdx.x;        // 0..31
    const int half = lane >> 4;          // 0 or 1 (lane half)
    const int lm   = lane & 15;
    const int tm   = blockIdx.x << 4;                                  // row tile base
    const int tn   = ((blockIdx.y * blockDim.y) + threadIdx.y) << 4;   // col tile base

    v8f acc = {};
    // A 16x4 f32 frag: VGPR0 holds K = k0 + half*2, VGPR1 holds K = k0 + half*2 + 1,
    // lanes 0..15 / 16..31 both span M = 0..15 of the tile.
    const float* ap = A + (size_t)(tm + lm) * K + (half << 1);
    // B 4x16 frag: lanes span N, VGPRs span K within the lane-half's K pair.
    const float* bp = B + (size_t)(half << 1) * NO + tn + lm;

    for (int k0 = 0; k0 < K; k0 += 4) {
        float a0 = ap[k0];
        float a1 = ap[k0 + 1];
        if (RELU) { a0 = fmaxf(a0, 0.f); a1 = fmaxf(a1, 0.f); }
        v2f a; a.x = a0; a.y = a1;
        v2f b; b.x = bp[(size_t)k0 * NO]; b.y = bp[(size_t)(k0 + 1) * NO];
        acc = __builtin_amdgcn_wmma_f32_16x16x4_f32(false, a, false, b,
                                                    (short)0, acc, false, false);
    }
    // C/D 16x16 f32: VGPR v -> row (v + half*8), lanes span columns.
    float* cp = C + (size_t)(tm + half * 8) * NO + tn + lm;
#pragma unroll
    for (int v = 0; v < 8; ++v) cp[(size_t)v * NO] = acc[v];
}

// 32 lanes per edge; each lane moves float4 of the 128-wide feature row.
__global__ void k_scatter(const float* __restrict__ h, const int* __restrict__ src,
                          const int* __restrict__ dst, const float* __restrict__ w,
                          float* __restrict__ agg, int E) {
    long long idx = (long long)blockIdx.x * blockDim.x + threadIdx.x;
    int e = (int)(idx >> 5);
    if (e >= E) return;
    int c = ((int)idx & 31) << 2;
    int s = src[e], d = dst[e];
    float ww = w[e];
    const float4 v = *(const float4*)(h + (size_t)s * HID + c);
    float* p = agg + (size_t)d * HID + c;
    atomicAdd(p + 0, v.x * ww);
    atomicAdd(p + 1, v.y * ww);
    atomicAdd(p + 2, v.z * ww);
    atomicAdd(p + 3, v.w * ww);
}

__global__ void k_zero16(float* __restrict__ diag) {
    diag[threadIdx.x] = 0.f;
}

// Per node: s = relu(agg2_row) @ Wm + bm ; softmax ; write sm ; accumulate diag += sm^2
__global__ void k_head(const float* __restrict__ agg2, const float* __restrict__ Wm,
                       const float* __restrict__ bm, float* __restrict__ out,
                       float* __restrict__ diag, int N) {
    __shared__ float sdiag[KCL];
    const int t = threadIdx.x;
    if (t < KCL) sdiag[t] = 0.f;
    __syncthreads();

    const int i = blockIdx.x * blockDim.x + t;
    if (i < N) {
        float acc[KCL];
#pragma unroll
        for (int j = 0; j < KCL; ++j) acc[j] = bm[j];
        const float* row = agg2 + (size_t)i * HID;
        for (int k = 0; k < HID; ++k) {
            float a = fmaxf(row[k], 0.f);         // fused relu of layer-2 aggregation
            const float* wr = Wm + k * KCL;
#pragma unroll
            for (int j = 0; j < KCL; ++j) acc[j] = fmaf(a, wr[j], acc[j]);
        }
        float m = acc[0];
#pragma unroll
        for (int j = 1; j < KCL; ++j) m = fmaxf(m, acc[j]);
        float p[KCL];
        float s = 0.f;
#pragma unroll
        for (int j = 0; j < KCL; ++j) { p[j] = __expf(acc[j] - m); s += p[j]; }
        const float inv = 1.f / s;
        float* op = out + (size_t)i * KCL;
#pragma unroll
        for (int j = 0; j < KCL; ++j) {
            float pj = p[j] * inv;
            op[j] = pj;
            atomicAdd(&sdiag[j], pj * pj);        // LDS (ds_add_f32) partial reduction
        }
    }
    __syncthreads();
    if (t < KCL) atomicAdd(&diag[t], sdiag[t]);
}

__global__ void k_finalize(const float* __restrict__ diag, float* __restrict__ loss, int N) {
    float s = 0.f;
#pragma unroll
    for (int j = 0; j < KCL; ++j) s += sqrtf(diag[j] + 1e-15f);
    *loss = -s / sqrtf((float)N * (float)KCL);
}

extern "C" void kernel_launch(void* const* d_in, const int* in_sizes, int n_in,
                              void* d_out, int out_size, void* d_ws, size_t ws_size,
                              hipStream_t stream) {
    const float* x  = (const float*)d_in[0];
    const int*   ei = (const int*)  d_in[1];
    const float* ew = (const float*)d_in[2];
    const float* W1 = (const float*)d_in[3];
    const float* b1 = (const float*)d_in[4];
    const float* W2 = (const float*)d_in[5];
    const float* b2 = (const float*)d_in[6];
    const float* Wm = (const float*)d_in[7];
    const float* bm = (const float*)d_in[8];

    const int N = in_sizes[0] / IN_F;   // 50000
    const int E = in_sizes[2];          // 800000
    const int* src = ei;
    const int* dst = ei + E;
    float* out = (float*)d_out;

    float* buf0 = (float*)d_ws;                 // hx / hy  [N,128]
    float* buf1 = buf0 + (size_t)N * HID;       // agg1 / agg2 [N,128]
    float* diag = buf1 + (size_t)N * HID;       // [16]

    const int totNH = N * HID;
    const dim3 gGemm(N / 16, HID / 64), bGemm(32, 4);
    const unsigned sBlocks = (unsigned)(((long long)E * 32 + 255) / 256);

    // Layer 1: agg1 = b1 ; hx = x @ W1 ; scatter w*hx[src] -> agg1[dst]
    k_init_bias<<<(totNH + 255) / 256, 256, 0, stream>>>(buf1, b1, totNH);
    k_gemm_wmma<false><<<gGemm, bGemm, 0, stream>>>(x, W1, buf0, IN_F, HID);
    k_scatter<<<sBlocks, 256, 0, stream>>>(buf0, src, dst, ew, buf1, E);

    // Layer 2: hy = relu(agg1) @ W2 ; agg2 = b2 ; scatter w*hy[src] -> agg2[dst]
    k_gemm_wmma<true><<<gGemm, bGemm, 0, stream>>>(buf1, W2, buf0, HID, HID);
    k_init_bias<<<(totNH + 255) / 256, 256, 0, stream>>>(buf1, b2, totNH);
    k_scatter<<<sBlocks, 256, 0, stream>>>(buf0, src, dst, ew, buf1, E);

    // Head: softmax(relu(agg2) @ Wm + bm), diag accumulation, loss
    k_zero16<<<1, KCL, 0, stream>>>(diag);
    k_head<<<(N + 255) / 256, 256, 0, stream>>>(buf1, Wm, bm, out, diag, N);
    k_finalize<<<1, 1, 0, stream>>>(diag, out + (size_t)N * KCL, N);
}